// CrossAttention_39436389712000
// MI455X (gfx1250) — compile-verified
//
#include <hip/hip_runtime.h>
#include <math.h>

// Problem constants (fixed by the reference)
#define BB 4
#define CC 64
#define HH 64
#define WW 64
#define NN (HH * WW)   // 4096 spatial positions
#define RR 16          // reduced channels == WMMA tile dim
#define LN_EPS 1e-5f

typedef float v2f __attribute__((ext_vector_type(2)));
typedef float v8f __attribute__((ext_vector_type(8)));

// ---------------------------------------------------------------------------
// Kernel 1: QKV projections (1x1 convs).  One thread per (b, n) pixel.
// Q/K packed as [b][n][r] (16 floats contiguous) for WMMA B/A fragment loads;
// V packed transposed as [b][r][n] so PV A-fragments are contiguous float2s.
// ---------------------------------------------------------------------------
__global__ __launch_bounds__(256) void proj_kernel(
    const float* __restrict__ x, const float* __restrict__ y,
    const float* __restrict__ Wq, const float* __restrict__ bq,
    const float* __restrict__ Wk, const float* __restrict__ bk,
    const float* __restrict__ Wv, const float* __restrict__ bv,
    float* __restrict__ Qb, float* __restrict__ Kb, float* __restrict__ Vt) {
  int t = blockIdx.x * 256 + threadIdx.x;        // 0 .. B*N-1
  int b = t / NN, n = t % NN;
  const float* xb = x + (size_t)b * CC * NN + n; // stride NN over c
  const float* yb = y + (size_t)b * CC * NN + n;

  float q[RR], k[RR], v[RR];
#pragma unroll
  for (int r = 0; r < RR; ++r) { q[r] = bq[r]; k[r] = bk[r]; v[r] = bv[r]; }

#pragma unroll 4
  for (int c = 0; c < CC; ++c) {
    float xv = xb[(size_t)c * NN];
    float yv = yb[(size_t)c * NN];
#pragma unroll
    for (int r = 0; r < RR; ++r) {
      q[r] = fmaf(Wq[r * CC + c], xv, q[r]);
      k[r] = fmaf(Wk[r * CC + c], yv, k[r]);
      v[r] = fmaf(Wv[r * CC + c], yv, v[r]);
    }
  }
  float* qo = Qb + (size_t)t * RR;
  float* ko = Kb + (size_t)t * RR;
  float* vo = Vt + (size_t)b * RR * NN + n;      // transposed: [b][r][n]
#pragma unroll
  for (int r = 0; r < RR; ++r) {
    qo[r] = q[r];
    ko[r] = k[r];
    vo[(size_t)r * NN] = v[r];                   // coalesced across threads
  }
}

// ---------------------------------------------------------------------------
// Kernel 2: flash attention.  8 waves / block, each wave owns one tile of 16
// queries; streams over 256 key tiles with online softmax.
//   Sᵀ tile = K·Qᵀ   (v_wmma_f32_16x16x4_f32 x4, chained over dim)
//   PV      = V·Pᵀ   (v_wmma_f32_16x16x4_f32 x4, accumulating in registers)
// Pᵀ bounces through a padded per-wave LDS tile to convert C-layout fragments
// into B-layout fragments.
// ---------------------------------------------------------------------------
__global__ __launch_bounds__(256) void attn_kernel(
    const float* __restrict__ Qb, const float* __restrict__ Kb,
    const float* __restrict__ Vt, float* __restrict__ Ob) {
  __shared__ float lds[8][16 * 17];              // stride 17: no bank conflicts

  const int lane = threadIdx.x & 31;
  const int wave = threadIdx.x >> 5;
  const int tile = blockIdx.x * 8 + wave;        // 0 .. B*N/16 - 1
  const int b = tile >> 8;                       // 256 query tiles per batch
  const int n0 = (tile & 255) * 16;
  const int col = lane & 15;                     // query column for this lane
  const int off = (lane < 16) ? 0 : 2;           // K-offset per ISA A/B layout
  const int rowbase = (lane < 16) ? 0 : 8;       // C/D layout row offset

  // Loop-invariant B-fragments of Qᵀ (rows = dim, cols = query)
  const float* qp = Qb + ((size_t)(b * NN + n0) + col) * RR;
  v2f fq[4];
#pragma unroll
  for (int t = 0; t < 4; ++t) fq[t] = *(const v2f*)(qp + 4 * t + off);

  v8f acc = {0.f, 0.f, 0.f, 0.f, 0.f, 0.f, 0.f, 0.f};
  float rmax = -INFINITY, rsum = 0.f;
  float* lp = &lds[wave][0];

  // Per-lane base for V A-fragments: row r = col, contiguous over key index.
  const float* vrow = Vt + ((size_t)b * RR + col) * NN;

  for (int j = 0; j < NN / 16; ++j) {
    // ---- Sᵀ = K · Qᵀ  (M = key, N = query) ----
    const float* kp = Kb + ((size_t)(b * NN + j * 16) + col) * RR;
    v2f fk[4];
#pragma unroll
    for (int t = 0; t < 4; ++t) fk[t] = *(const v2f*)(kp + 4 * t + off);

    v8f s = {0.f, 0.f, 0.f, 0.f, 0.f, 0.f, 0.f, 0.f};
#pragma unroll
    for (int t = 0; t < 4; ++t)
      s = __builtin_amdgcn_wmma_f32_16x16x4_f32(false, fk[t], false, fq[t],
                                                (short)0, s, false, false);

    // ---- online softmax over keys (columns of Sᵀ) ----
    float m = s[0];
#pragma unroll
    for (int i = 1; i < 8; ++i) m = fmaxf(m, s[i]);
    m = fmaxf(m, __shfl_xor(m, 16, 32));         // combine lane halves
    float mnew = fmaxf(rmax, m);
    float scale = __expf(rmax - mnew);

    float p[8], lsum = 0.f;
#pragma unroll
    for (int i = 0; i < 8; ++i) { p[i] = __expf(s[i] - mnew); lsum += p[i]; }
    lsum += __shfl_xor(lsum, 16, 32);
    rsum = rsum * scale + lsum;
    rmax = mnew;

    // ---- Pᵀ: C-layout regs -> LDS -> B-layout fragments ----
#pragma unroll
    for (int i = 0; i < 8; ++i) lp[(i + rowbase) * 17 + col] = p[i];
    asm volatile("s_wait_dscnt 0" ::: "memory"); // same-wave LDS RAW fence

#pragma unroll
    for (int i = 0; i < 8; ++i) acc[i] *= scale;

    const float* vp = vrow + j * 16;             // keys contiguous per lane
    v2f fv[4], fp[4];
#pragma unroll
    for (int t = 0; t < 4; ++t) {
      fv[t] = *(const v2f*)(vp + 4 * t + off);   // A = V (M = dim, K = key)
      fp[t].x = lp[(4 * t + off) * 17 + col];    // B = Pᵀ (rows = key)
      fp[t].y = lp[(4 * t + off + 1) * 17 + col];
    }
#pragma unroll
    for (int t = 0; t < 4; ++t)
      acc = __builtin_amdgcn_wmma_f32_16x16x4_f32(false, fv[t], false, fp[t],
                                                  (short)0, acc, false, false);
  }

  // ---- epilogue: O[b][n][r] = acc / rsum ----
  float inv = 1.0f / rsum;
  float* op = Ob + ((size_t)(b * NN + n0) + col) * RR + rowbase;
#pragma unroll
  for (int i = 0; i < 8; ++i) op[i] = acc[i] * inv;
}

// ---------------------------------------------------------------------------
// Kernel 3: final 1x1 conv + residual, writes pre-LN value into d_out and
// accumulates per-batch sum / sum-of-squares for LayerNorm.
// ---------------------------------------------------------------------------
__global__ __launch_bounds__(256) void fuse_kernel(
    const float* __restrict__ x, const float* __restrict__ Wf,
    const float* __restrict__ bf, const float* __restrict__ Ob,
    float* __restrict__ out, float* __restrict__ stats) {
  __shared__ float s1[256], s2[256];
  int idx = blockIdx.x * 256 + threadIdx.x;      // over B*C*N
  int b = idx / (CC * NN);
  int cn = idx % (CC * NN);
  int c = cn / NN, n = cn % NN;

  const float* op = Ob + (size_t)(b * NN + n) * RR;
  float a = bf[c];
#pragma unroll
  for (int r = 0; r < RR; ++r) a = fmaf(Wf[c * RR + r], op[r], a);
  a += x[idx];
  out[idx] = a;

  s1[threadIdx.x] = a;
  s2[threadIdx.x] = a * a;
  __syncthreads();
  for (int w = 128; w > 0; w >>= 1) {
    if (threadIdx.x < w) {
      s1[threadIdx.x] += s1[threadIdx.x + w];
      s2[threadIdx.x] += s2[threadIdx.x + w];
    }
    __syncthreads();
  }
  if (threadIdx.x == 0) {
    atomicAdd(&stats[b * 2 + 0], s1[0]);
    atomicAdd(&stats[b * 2 + 1], s2[0]);
  }
}

__global__ void init_stats_kernel(float* stats) {
  if (threadIdx.x < 16) stats[threadIdx.x] = 0.f;
}

__global__ void stats_final_kernel(const float* stats, float* ms) {
  int t = threadIdx.x;
  if (t < BB) {
    const float cnt = (float)(CC * NN);
    float mean = stats[t * 2 + 0] / cnt;
    float var = stats[t * 2 + 1] / cnt - mean * mean;
    ms[t * 2 + 0] = mean;
    ms[t * 2 + 1] = rsqrtf(var + LN_EPS);
  }
}

__global__ __launch_bounds__(256) void ln_kernel(
    const float* __restrict__ ln_w, const float* __restrict__ ln_b,
    const float* __restrict__ ms, float* __restrict__ out) {
  int idx = blockIdx.x * 256 + threadIdx.x;
  int b = idx / (CC * NN);
  int cn = idx % (CC * NN);
  float v = out[idx];
  out[idx] = (v - ms[b * 2 + 0]) * ms[b * 2 + 1] * ln_w[cn] + ln_b[cn];
}

// ---------------------------------------------------------------------------
extern "C" void kernel_launch(void* const* d_in, const int* in_sizes, int n_in,
                              void* d_out, int out_size, void* d_ws, size_t ws_size,
                              hipStream_t stream) {
  const float* x   = (const float*)d_in[0];
  const float* y   = (const float*)d_in[1];
  const float* Wq  = (const float*)d_in[2];
  const float* bq  = (const float*)d_in[3];
  const float* Wk  = (const float*)d_in[4];
  const float* bk  = (const float*)d_in[5];
  const float* Wv  = (const float*)d_in[6];
  const float* bv  = (const float*)d_in[7];
  const float* Wf  = (const float*)d_in[8];
  const float* bf  = (const float*)d_in[9];
  const float* lnw = (const float*)d_in[10];
  const float* lnb = (const float*)d_in[11];
  float* out = (float*)d_out;

  // Workspace layout (floats): Q | K | Vt | O | stats | mean/rsig
  float* ws = (float*)d_ws;
  const size_t qkv = (size_t)BB * NN * RR;       // 262144 floats each
  float* Qb = ws;
  float* Kb = Qb + qkv;
  float* Vt = Kb + qkv;
  float* Ob = Vt + qkv;
  float* stats = Ob + qkv;                        // 2*B floats
  float* ms = stats + 16;                         // 2*B floats

  init_stats_kernel<<<1, 32, 0, stream>>>(stats);

  proj_kernel<<<(BB * NN) / 256, 256, 0, stream>>>(x, y, Wq, bq, Wk, bk, Wv, bv,
                                                   Qb, Kb, Vt);

  attn_kernel<<<(BB * NN / 16) / 8, 256, 0, stream>>>(Qb, Kb, Vt, Ob);

  fuse_kernel<<<(BB * CC * NN) / 256, 256, 0, stream>>>(x, Wf, bf, Ob, out, stats);

  stats_final_kernel<<<1, 32, 0, stream>>>(stats, ms);

  ln_kernel<<<(BB * CC * NN) / 256, 256, 0, stream>>>(lnw, lnb, ms, out);
}